// CPCLoss_84378927497689
// MI455X (gfx1250) — compile-verified
//
#include <hip/hip_runtime.h>
#include <math.h>

typedef __attribute__((ext_vector_type(2))) float v2f;
typedef __attribute__((ext_vector_type(8))) float v8f;

#define EPS_CONST 1e-7f
#define WAVES_PER_BLOCK 8
#define BLOCK_THREADS (WAVES_PER_BLOCK * 32)
#define MAX_PAD 128    // padded row length (C=100 -> pad=112)
#define MAX_TILES 8

#define LOG2E 1.4426950408889634f
#define LN2   0.6931471805599453f
#define BIG_SENTINEL 1.0e4f   // logsig(+BIG - r + eps) == 0.0f exactly in f32

// Fast, branch-free log(sigmoid(x)) using raw v_exp_f32 / v_log_f32:
//   logsig(x) = min(x,0) - ln2 * log2(1 + 2^(-|x|*log2e))
// Argument of log2 is in (1,2], so no denormal/infinity fixups are needed.
__device__ __forceinline__ float logsig(float x) {
    float e = __builtin_amdgcn_exp2f(-fabsf(x) * LOG2E);
    float l = __builtin_amdgcn_logf(1.0f + e);
    return fminf(x, 0.0f) - LN2 * l;
}

__device__ __forceinline__ float wave_reduce_sum(float v) {
#pragma unroll
    for (int m = 16; m >= 1; m >>= 1) v += __shfl_xor(v, m, 32);
    return v;
}

__device__ __forceinline__ float wave_reduce_max(float v) {
#pragma unroll
    for (int m = 16; m >= 1; m >>= 1) v = fmaxf(v, __shfl_xor(v, m, 32));
    return v;
}

// TILES_CT > 0: compile-time tile count (loops fully unrolled).
// TILES_CT == 0: runtime tile count fallback.
template <int TILES_CT>
__global__ void __launch_bounds__(BLOCK_THREADS)
cpc_main(const float* __restrict__ x, const int* __restrict__ tgt,
         float* __restrict__ partials, int n, int c) {
    __shared__ float lrow[WAVES_PER_BLOCK][MAX_PAD];
    __shared__ float blk[3][WAVES_PER_BLOCK];

    const int lane  = threadIdx.x & 31;
    const int wave  = threadIdx.x >> 5;
    const int tiles = TILES_CT ? TILES_CT : ((c + 15) >> 4);
    const int pad   = tiles << 4;

    float ce = 0.0f, bdc = 0.0f, bec = 0.0f;

    for (int row = blockIdx.x * WAVES_PER_BLOCK + wave; row < n;
         row += gridDim.x * WAVES_PER_BLOCK) {
        const float* xr = x + (size_t)row * (size_t)c;
        const int t = tgt[row];

        // Stage row into LDS (zero-padded to tiles*16).
        for (int j = lane; j < pad; j += 32)
            lrow[wave][j] = (j < c) ? xr[j] : 0.0f;
        // Same-wave LDS ops are in-order; wait for the fill before gathers.
        asm volatile("s_wait_dscnt 0x0" ::: "memory");

        // ---- Cross entropy: logsumexp over the row (base-2 domain) ----
        float m = -INFINITY;
        for (int j = lane; j < c; j += 32) m = fmaxf(m, lrow[wave][j]);
        m = wave_reduce_max(m);
        float se = 0.0f;
        for (int j = lane; j < c; j += 32)
            se += __builtin_amdgcn_exp2f((lrow[wave][j] - m) * LOG2E);
        se = wave_reduce_sum(se);
        const float xt = lrow[wave][t];
        const float lse = m + LN2 * __builtin_amdgcn_logf(se);
        ce += (lane == 0) ? (lse - xt) : 0.0f;        // -log p_t

        // ---- BDC: sum_{j!=t} logsigmoid(x_t - x_j), branch-free ----
        for (int j = lane; j < c; j += 32) {
            float l = logsig(xt - lrow[wave][j]);
            bdc += (j != t) ? l : 0.0f;
        }

        // ---- BEC: pairwise diffs via rank-2 WMMA with sentinel masking ----
        // Invalid m-rows get +BIG, invalid n-cols get -BIG, so every invalid
        // pair yields D = +huge and logsig(D) == 0.0f exactly: the inner loop
        // needs no masks and no branches.
        const int  l15 = lane & 15;
        const bool lo  = lane < 16;

        float bvs[TILES_CT ? TILES_CT : MAX_TILES];
#pragma unroll
        for (int tj = 0; tj < tiles; ++tj) {
            const int   idx = tj * 16 + l15;
            const float rv  = lrow[wave][idx];
            const bool  ok  = (idx < c) && (idx != t);
            bvs[tj] = ok ? rv : -BIG_SENTINEL;
        }

        for (int ti = 0; ti < tiles; ++ti) {
            const int   aIdx = ti * 16 + l15;
            const float arv  = lrow[wave][aIdx];
            const bool  aOk  = (aIdx < c) && (aIdx != t);
            const float av   = aOk ? arv : BIG_SENTINEL;
            v2f A;
            A.x = lo ? av    : 0.0f;   // K=0 column: masked r_m
            A.y = lo ? -1.0f : 0.0f;   // K=1 column: -1
#pragma unroll
            for (int tj = 0; tj < tiles; ++tj) {
                v2f B;
                B.x = lo ? 1.0f    : 0.0f; // K=0 row: ones
                B.y = lo ? bvs[tj] : 0.0f; // K=1 row: masked r_n
                v8f Cm;
#pragma unroll
                for (int v = 0; v < 8; ++v) Cm[v] = EPS_CONST;
                // D[m][n] = a_m - b_n + eps (16x16 tile per matrix op)
                v8f D = __builtin_amdgcn_wmma_f32_16x16x4_f32(
                    false, A, false, B, (short)0, Cm, false, false);
#pragma unroll
                for (int v = 0; v < 8; ++v) bec += logsig(D[v]);
            }
        }
    }

    ce  = wave_reduce_sum(ce);
    bdc = wave_reduce_sum(bdc);
    bec = wave_reduce_sum(bec);
    if (lane == 0) {
        blk[0][wave] = ce; blk[1][wave] = bdc; blk[2][wave] = bec;
    }
    __syncthreads();
    if (threadIdx.x == 0) {
        float s0 = 0.0f, s1 = 0.0f, s2 = 0.0f;
        for (int w = 0; w < WAVES_PER_BLOCK; ++w) {
            s0 += blk[0][w]; s1 += blk[1][w]; s2 += blk[2][w];
        }
        partials[3 * blockIdx.x + 0] = s0;
        partials[3 * blockIdx.x + 1] = s1;
        partials[3 * blockIdx.x + 2] = s2;
    }
}

__global__ void __launch_bounds__(256)
cpc_final(const float* __restrict__ partials, float* __restrict__ out,
          int nblocks, int n, int c) {
    __shared__ float s[3][256];
    float a = 0.0f, b = 0.0f, d = 0.0f;
    for (int i = threadIdx.x; i < nblocks; i += 256) {
        a += partials[3 * i + 0];
        b += partials[3 * i + 1];
        d += partials[3 * i + 2];
    }
    s[0][threadIdx.x] = a; s[1][threadIdx.x] = b; s[2][threadIdx.x] = d;
    __syncthreads();
    for (int off = 128; off >= 1; off >>= 1) {
        if (threadIdx.x < off) {
            s[0][threadIdx.x] += s[0][threadIdx.x + off];
            s[1][threadIdx.x] += s[1][threadIdx.x + off];
            s[2][threadIdx.x] += s[2][threadIdx.x + off];
        }
        __syncthreads();
    }
    if (threadIdx.x == 0) {
        const float fn = (float)n;
        const float loss_ce  = s[0][0] / fn;
        const float loss_bdc = -s[1][0] / (float)(c - 1) / fn;
        const float loss_bec = -0.5f * s[2][0] / (float)(c - 1) / (float)(c - 2) / fn;
        out[0] = loss_ce + loss_bdc + loss_bec; // LAMBD_BDC = LAMBD_BEC = 1
        out[1] = loss_ce;
        out[2] = loss_bdc;
        out[3] = loss_bec;
    }
}

extern "C" void kernel_launch(void* const* d_in, const int* in_sizes, int n_in,
                              void* d_out, int out_size, void* d_ws, size_t ws_size,
                              hipStream_t stream) {
    (void)n_in; (void)out_size;
    const float* x   = (const float*)d_in[0];
    const int*   tgt = (const int*)d_in[1];
    const int n = in_sizes[1];              // 8192 targets
    const int c = in_sizes[0] / n;          // 100 classes
    float* out      = (float*)d_out;        // 4 floats
    float* partials = (float*)d_ws;

    int blocks = 1024;                      // 8 waves/block -> 1 row per wave
    const int maxBlocks = (int)(ws_size / (3 * sizeof(float)));
    if (blocks > maxBlocks) blocks = maxBlocks > 0 ? maxBlocks : 1;

    const int tiles = (c + 15) >> 4;
    if (tiles == 7) {                       // C == 97..112 (reference: 100)
        hipLaunchKernelGGL((cpc_main<7>), dim3(blocks), dim3(BLOCK_THREADS), 0,
                           stream, x, tgt, partials, n, c);
    } else {                                // generic runtime-tiles fallback
        hipLaunchKernelGGL((cpc_main<0>), dim3(blocks), dim3(BLOCK_THREADS), 0,
                           stream, x, tgt, partials, n, c);
    }
    hipLaunchKernelGGL(cpc_final, dim3(1), dim3(256), 0, stream,
                       partials, out, blocks, n, c);
}